// Model_41472204210267
// MI455X (gfx1250) — compile-verified
//
#include <hip/hip_runtime.h>

// Problem constants (from reference: V=8192, B=8, T=1024, C==V)
#define VOCAB   8192
#define BATCH   8
#define SEQ     1024
#define CW      128                 // channels per workgroup (8 waves * 16)
#define NCHUNK  (VOCAB / CW)        // 64 channel chunks
#define NWAVES  8
#define WGSIZE  256

typedef float v2f __attribute__((ext_vector_type(2)));
typedef float v8f __attribute__((ext_vector_type(8)));

// ---------------------------------------------------------------------------
// Kernel A: fused gather + logits write + causal prefix (WMMA) + chunk stats
// grid = BATCH * NCHUNK workgroups, 256 threads each.
// Each wave owns a 16-channel column; per 16-timestep tile it computes
//   D = tril(16x16) @ X + broadcast(carry)        (4x V_WMMA_F32_16X16X4_F32)
// giving the causal running sums S for 16 timesteps at once.
// ---------------------------------------------------------------------------
__global__ __launch_bounds__(WGSIZE)
void gather_prefix_stats(const int* __restrict__ x,
                         const int* __restrict__ y,
                         const float* __restrict__ emb,
                         float* __restrict__ logits,
                         float* __restrict__ ws_max,
                         float* __restrict__ ws_sum,
                         float* __restrict__ ws_my)
{
    const int wg    = blockIdx.x;
    const int b     = wg / NCHUNK;
    const int chunk = wg % NCHUNK;
    const int c0    = chunk * CW;
    const int tid   = threadIdx.x;
    const int wave  = tid >> 5;       // wave32
    const int lane  = tid & 31;
    const int half  = lane >> 4;      // 0: lanes 0-15, 1: lanes 16-31
    const int ln    = lane & 15;
    const int c     = c0 + wave * 16 + ln;   // this lane's channel column

    __shared__ __align__(16) int   s_x[16];
    __shared__ __align__(16) int   s_y[16];
    __shared__ float s_red[16][NWAVES];
    __shared__ float s_row[16];

    // A-matrix fragments: L = tril(ones(16,16)) split into four 16x4 K-blocks.
    // 16x4 f32 A layout: lanes 0-15 -> M=0..15; VGPR0 = K=(0|2), VGPR1 = K=(1|3)
    v2f A[4];
    {
        const int m = ln;
        #pragma unroll
        for (int k = 0; k < 4; ++k) {
            const int kk = 4 * k + half * 2;
            A[k].x = (m >= kk    ) ? 1.0f : 0.0f;
            A[k].y = (m >= kk + 1) ? 1.0f : 0.0f;
        }
    }

    float carry = 0.0f;   // running sum of logits[b, 0..t0-1, c]

    for (int t0 = 0; t0 < SEQ; t0 += 16) {
        if (tid < 16) {
            s_x[tid] = x[b * SEQ + t0 + tid];
            s_y[tid] = y[b * SEQ + t0 + tid];
        }
        __syncthreads();

        // Prefetch start of next tile's gathered rows (global_prefetch_b8).
        if (tid < 16 && (t0 + 16) < SEQ) {
            const int xn = x[b * SEQ + t0 + 16 + tid];
            __builtin_prefetch(&emb[(long)xn * VOCAB + c0], 0, 0);
        }

        // Pull this tile's x / y values out of LDS once, with wide ds loads,
        // so the inner compare/store loops are register-only (no per-row
        // ds_load + s_wait_dscnt chains).
        int2 xp[4];
        #pragma unroll
        for (int k = 0; k < 4; ++k)
            xp[k] = *(const int2*)&s_x[4 * k + half * 2];   // rows s0, s0+1

        int yv[8];
        {
            const int4 ya = *(const int4*)&s_y[half * 8];
            const int4 yb = *(const int4*)&s_y[half * 8 + 4];
            yv[0] = ya.x; yv[1] = ya.y; yv[2] = ya.z; yv[3] = ya.w;
            yv[4] = yb.x; yv[5] = yb.y; yv[6] = yb.z; yv[7] = yb.w;
        }

        // B-matrix fragments: X[s][c] = emb[x[b,t0+s], c], 4x16 K-blocks.
        // 4x16 f32 B layout (mirror of A): lanes 0-15 -> N; VGPR0=K(0|2), VGPR1=K(1|3)
        v2f Bf[4];
        #pragma unroll
        for (int k = 0; k < 4; ++k) {
            const int s0 = 4 * k + half * 2;
            const float x0 = emb[(long)xp[k].x * VOCAB + c];
            const float x1 = emb[(long)xp[k].y * VOCAB + c];
            Bf[k].x = x0;
            Bf[k].y = x1;
            logits[((long)(b * SEQ + t0 + s0    )) * VOCAB + c] = x0;
            logits[((long)(b * SEQ + t0 + s0 + 1)) * VOCAB + c] = x1;
        }

        // D = L @ X + C, with C = carry broadcast down all 16 rows.
        v8f D = { carry, carry, carry, carry, carry, carry, carry, carry };
        #pragma unroll
        for (int k = 0; k < 4; ++k) {
            D = __builtin_amdgcn_wmma_f32_16x16x4_f32(
                    false, A[k], false, Bf[k], (short)0, D, false, false);
        }

        // Per-lane rows r = v + half*8; scale to causal mean m = S/(t+1).
        // v_rcp_f32 (1 TRANS op, co-executes with the WMMA chain) instead of
        // the IEEE divide sequence; divisor is an exact small integer and the
        // result only feeds a softmax, so approx rcp is fine.
        float e[8];
        #pragma unroll
        for (int v = 0; v < 8; ++v) {
            const int r = v + half * 8;
            const float inv = __builtin_amdgcn_rcpf((float)(t0 + r + 1));
            e[v] = D[v] * inv;
            if (yv[v] == c) ws_my[b * SEQ + t0 + r] = e[v];
        }

        // New carry = running sum through row 15 = D[7] of lanes 16..31.
        carry = __shfl(D[7], 16 + ln, 32);

        // Row-wise max over this wave's 16 channels (xor shuffles stay in half).
        #pragma unroll
        for (int v = 0; v < 8; ++v) {
            float m = e[v];
            m = fmaxf(m, __shfl_xor(m, 1, 32));
            m = fmaxf(m, __shfl_xor(m, 2, 32));
            m = fmaxf(m, __shfl_xor(m, 4, 32));
            m = fmaxf(m, __shfl_xor(m, 8, 32));
            if (ln == 0) s_red[v + half * 8][wave] = m;
        }
        __syncthreads();

        if (tid < 16) {
            float m = s_red[tid][0];
            #pragma unroll
            for (int w = 1; w < NWAVES; ++w) m = fmaxf(m, s_red[tid][w]);
            s_row[tid] = m;
        }
        __syncthreads();

        // Row-wise sum of exp(m - rowmax) over the chunk.
        #pragma unroll
        for (int v = 0; v < 8; ++v) {
            const int r = v + half * 8;
            float s = __expf(e[v] - s_row[r]);
            s += __shfl_xor(s, 1, 32);
            s += __shfl_xor(s, 2, 32);
            s += __shfl_xor(s, 4, 32);
            s += __shfl_xor(s, 8, 32);
            if (ln == 0) s_red[r][wave] = s;
        }
        __syncthreads();

        if (tid < 16) {
            float s = 0.0f;
            #pragma unroll
            for (int w = 0; w < NWAVES; ++w) s += s_red[tid][w];
            const long pos = (long)(b * SEQ + t0 + tid);
            ws_max[pos * NCHUNK + chunk] = s_row[tid];
            ws_sum[pos * NCHUNK + chunk] = s;
        }
        __syncthreads();   // protect s_x/s_y/s_red before next tile
    }
}

// ---------------------------------------------------------------------------
// Kernel B: per-position logsumexp combine over the 64 chunks -> per-WG partial
// ---------------------------------------------------------------------------
__global__ __launch_bounds__(WGSIZE)
void combine_stats(const float* __restrict__ ws_max,
                   const float* __restrict__ ws_sum,
                   const float* __restrict__ ws_my,
                   float* __restrict__ ws_partial)
{
    const int pos = blockIdx.x * WGSIZE + threadIdx.x;   // 0 .. B*T-1
    const float* mx = ws_max + (long)pos * NCHUNK;
    const float* sm = ws_sum + (long)pos * NCHUNK;

    float g = mx[0];
    #pragma unroll 8
    for (int i = 1; i < NCHUNK; ++i) g = fmaxf(g, mx[i]);
    float tot = 0.0f;
    #pragma unroll 8
    for (int i = 0; i < NCHUNK; ++i) tot += sm[i] * __expf(mx[i] - g);

    float nll = -(ws_my[pos] - g - __logf(tot));

    __shared__ float red[WGSIZE];
    red[threadIdx.x] = nll;
    __syncthreads();
    for (int s = WGSIZE / 2; s > 0; s >>= 1) {
        if (threadIdx.x < s) red[threadIdx.x] += red[threadIdx.x + s];
        __syncthreads();
    }
    if (threadIdx.x == 0) ws_partial[blockIdx.x] = red[0];
}

// ---------------------------------------------------------------------------
// Kernel C: deterministic final reduction -> loss scalar
// ---------------------------------------------------------------------------
__global__ void finalize_loss(const float* __restrict__ ws_partial,
                              float* __restrict__ loss, int nparts)
{
    if (threadIdx.x == 0) {
        float s = 0.0f;
        for (int i = 0; i < nparts; ++i) s += ws_partial[i];
        *loss = s / (float)(BATCH * SEQ);
    }
}

extern "C" void kernel_launch(void* const* d_in, const int* in_sizes, int n_in,
                              void* d_out, int out_size, void* d_ws, size_t ws_size,
                              hipStream_t stream)
{
    (void)in_sizes; (void)n_in; (void)out_size; (void)ws_size;

    const int*   x   = (const int*)d_in[0];
    const int*   y   = (const int*)d_in[1];
    const float* emb = (const float*)d_in[2];

    float* out    = (float*)d_out;
    float* logits = out;                                   // B*T*V floats
    float* loss   = out + (long)BATCH * SEQ * VOCAB;       // final scalar

    float* ws_max     = (float*)d_ws;                      // B*T*NCHUNK
    float* ws_sum     = ws_max + (long)BATCH * SEQ * NCHUNK;
    float* ws_my      = ws_sum + (long)BATCH * SEQ * NCHUNK;
    float* ws_partial = ws_my + (long)BATCH * SEQ;

    gather_prefix_stats<<<BATCH * NCHUNK, WGSIZE, 0, stream>>>(
        x, y, emb, logits, ws_max, ws_sum, ws_my);

    const int nparts = (BATCH * SEQ) / WGSIZE;             // 32
    combine_stats<<<nparts, WGSIZE, 0, stream>>>(ws_max, ws_sum, ws_my, ws_partial);
    finalize_loss<<<1, 32, 0, stream>>>(ws_partial, loss, nparts);
}